// DenseWarpLayer_30142080483698
// MI455X (gfx1250) — compile-verified
//
#include <hip/hip_runtime.h>

// dense_image_warp for image (4,720,1280,16) f32, flow (4,720,1280,2) f32.
// Bandwidth-bound gather: ~500 MB total traffic -> ~21.5 us floor @ 23.3 TB/s.
// One lane per (pixel, 4-channel quad): gathers coalesce into contiguous 64B
// runs per corner across lanes 0..3 of each pixel.

typedef __attribute__((ext_vector_type(4))) float v4f;

namespace {
constexpr int   kN  = 4;
constexpr int   kH  = 720;
constexpr int   kW  = 1280;
constexpr int   kC  = 16;           // channels
constexpr int   kHW = kH * kW;      // 921600
constexpr unsigned kPixels  = (unsigned)kN * kHW;          // 3,686,400
constexpr unsigned kThreads = kPixels * 4;                 // 14,745,600 (one per float4)
}

__global__ __launch_bounds__(256)
void dense_warp_kernel(const float* __restrict__ image,
                       const float* __restrict__ flow,
                       float* __restrict__ out)
{
    unsigned tid = blockIdx.x * blockDim.x + threadIdx.x;
    if (tid >= kThreads) return;

    const unsigned q = tid & 3u;        // which float4 of the 16 channels
    const unsigned p = tid >> 2;        // global pixel index: ((n*H)+i)*W + j

    // Decompose pixel index (constant divisors -> magic multiplies)
    const unsigned n   = p / (unsigned)kHW;
    const unsigned rem = p - n * (unsigned)kHW;
    const unsigned i   = rem / (unsigned)kW;
    const unsigned j   = rem - i * (unsigned)kW;

    // flow[..., 0] = dy, flow[..., 1] = dx  (contiguous pair per pixel)
    const float2 fl = ((const float2* __restrict__)flow)[p];

    const float qy = (float)i - fl.x;
    const float qx = (float)j - fl.y;

    // tfa _interpolate_bilinear semantics: floor clipped to [0, size-2],
    // alpha clipped to [0, 1].
    float fy = floorf(qy);
    float fx = floorf(qx);
    fy = fminf(fmaxf(fy, 0.0f), (float)(kH - 2));
    fx = fminf(fmaxf(fx, 0.0f), (float)(kW - 2));
    const float ay = fminf(fmaxf(qy - fy, 0.0f), 1.0f);
    const float ax = fminf(fmaxf(qx - fx, 0.0f), 1.0f);
    const int iy = (int)fy;
    const int ix = (int)fx;

    // Base index in float4 units: (((n*H + iy)*W + ix)*C)/4 + q
    const v4f* __restrict__ img = (const v4f* __restrict__)image;
    const unsigned rowBase = ((n * (unsigned)kH + (unsigned)iy) * (unsigned)kW
                              + (unsigned)ix) * 4u + q;

    const v4f tl = img[rowBase];
    const v4f tr = img[rowBase + 4u];                      // ix+1 -> +C floats
    const v4f bl = img[rowBase + (unsigned)kW * 4u];       // iy+1 -> +W*C floats
    const v4f br = img[rowBase + (unsigned)kW * 4u + 4u];

    const v4f top = tl + ax * (tr - tl);
    const v4f bot = bl + ax * (br - bl);
    const v4f res = top + ay * (bot - top);

    // Output is write-once / never re-read: non-temporal store keeps L2 for
    // the image working set (which neighboring pixels re-read).
    __builtin_nontemporal_store(res, ((v4f*)out) + (p * 4u + q));
}

extern "C" void kernel_launch(void* const* d_in, const int* in_sizes, int n_in,
                              void* d_out, int out_size, void* d_ws, size_t ws_size,
                              hipStream_t stream)
{
    (void)in_sizes; (void)n_in; (void)out_size; (void)d_ws; (void)ws_size;

    const float* image = (const float*)d_in[0];
    const float* flow  = (const float*)d_in[1];
    float*       out   = (float*)d_out;

    constexpr unsigned block = 256;
    constexpr unsigned grid  = (kThreads + block - 1) / block;   // 57600

    dense_warp_kernel<<<grid, block, 0, stream>>>(image, flow, out);
}